// RtrlCell_27350351741524
// MI455X (gfx1250) — compile-verified
//
#include <hip/hip_runtime.h>

typedef float v2f __attribute__((ext_vector_type(2)));
typedef float v8f __attribute__((ext_vector_type(8)));
typedef unsigned int u32x4 __attribute__((ext_vector_type(4)));
typedef int i32x8 __attribute__((ext_vector_type(8)));
typedef int i32x4 __attribute__((ext_vector_type(4)));

#define HDIM 128
#define BDIM 16
#define LSTRIDE 132   // padded LDS row stride (floats); TDM pad reproduces this

// ---------------------------------------------------------------------------
// Kernel 1: curr_h = tanh(x @ W + b + prev_h @ R)   [16,128] — trivial cost
// ---------------------------------------------------------------------------
__global__ __launch_bounds__(256) void rtrl_h_kernel(
    const float* __restrict__ x, const float* __restrict__ prev_h,
    const float* __restrict__ W, const float* __restrict__ bias,
    const float* __restrict__ R, float* __restrict__ out_h)
{
    for (int t = threadIdx.x; t < BDIM * HDIM; t += 256) {
        int b = t >> 7;
        int h = t & 127;
        const float* xr = x + b * HDIM;
        const float* pr = prev_h + b * HDIM;
        float s = bias[h];
        #pragma unroll 4
        for (int i = 0; i < HDIM; ++i) {
            s += xr[i] * W[i * HDIM + h];
            s += pr[i] * R[i * HDIM + h];
        }
        out_h[t] = tanhf(s);
    }
}

// ---------------------------------------------------------------------------
// Kernel 2: sensitivity recurrences via fp32 WMMA + TDM tile staging.
//   O[k][c] = sum_n A[k][n] * S[n][c] + diag*delta(k==c_local)
//   A[k][n] = R[n][k] * (1 - prev_h[b][n]^2)
// One block per (batch, {S_R tile | S_W tile | S_B}).
// The 128x128 S tile is DMA'd Global->LDS by the Tensor Data Mover with
// hardware padding (128 DW interval, 4 DW pad) == LSTRIDE layout.
// ---------------------------------------------------------------------------
__global__ __launch_bounds__(256) void rtrl_sens_kernel(
    const float* __restrict__ x, const float* __restrict__ prev_h,
    const float* __restrict__ S_R, const float* __restrict__ S_W,
    const float* __restrict__ S_B, const float* __restrict__ R,
    float* __restrict__ out)
{
    extern __shared__ float lds[];
    float* a_lds = lds;                    // [128][LSTRIDE]  A[n][k]
    float* s_lds = lds + HDIM * LSTRIDE;   // [128][LSTRIDE]  S rows (TDM dest)

    const int tid = threadIdx.x;
    const int per_batch = 2 * HDIM + 1;    // 128 S_R tiles + 128 S_W tiles + S_B
    const int b = blockIdx.x / per_batch;
    const int r = blockIdx.x % per_batch;

    const long long big = (long long)HDIM * HDIM * HDIM;   // 2097152 per batch
    const long long o_SR = (long long)BDIM * HDIM;         // 2048
    const long long o_SW = o_SR + (long long)BDIM * big;
    const long long o_SB = o_SW + (long long)BDIM * big;

    const float*  src;            // source S block: rows n=0..127
    long long     src_row_stride; // floats between consecutive n rows
    float*        ob;             // output block base (row k=0, col 0)
    long long     out_row_stride;
    float         diag;

    if (r < HDIM) {                         // S_R tile, i = r
        src = S_R + (long long)b * big + (long long)r * HDIM;
        src_row_stride = (long long)HDIM * HDIM;           // 16384
        ob = out + o_SR + (long long)b * big + (long long)r * HDIM;
        out_row_stride = (long long)HDIM * HDIM;
        diag = prev_h[b * HDIM + r];
    } else if (r < 2 * HDIM) {              // S_W tile, i = r-128
        int ct = r - HDIM;
        src = S_W + (long long)b * big + (long long)ct * HDIM;
        src_row_stride = (long long)HDIM * HDIM;
        ob = out + o_SW + (long long)b * big + (long long)ct * HDIM;
        out_row_stride = (long long)HDIM * HDIM;
        diag = x[b * HDIM + ct];
    } else {                                // S_B
        src = S_B + (long long)b * HDIM * HDIM;
        src_row_stride = HDIM;
        ob = out + o_SB + (long long)b * HDIM * HDIM;
        out_row_stride = HDIM;
        diag = 1.0f;
    }

    // ---- Issue TDM: DMA the 128x128 f32 tile Global->LDS with padding.
    // D# group0: count=1, lds_addr, 57-bit global_addr, type=2.
    // D# group1: data_size=2(4B), pad_enable, pad_interval=6(128DW),
    //            pad_amount=3(4DW), dims/tile=128, dim0_stride=row stride.
    if (tid < 32) {
        unsigned long long ga = (unsigned long long)(uintptr_t)src;
        unsigned lds_off = (unsigned)(uintptr_t)s_lds;   // addr[31:0] = LDS offset

        u32x4 g0;
        g0.x = 1u;                                        // count=1
        g0.y = lds_off;                                   // lds_addr
        g0.z = (unsigned)(ga & 0xFFFFFFFFu);              // global_addr[31:0]
        g0.w = (unsigned)((ga >> 32) & 0x01FFFFFFu) | 0x80000000u; // [56:32] | type=2

        i32x8 g1;
        g1[0] = (int)(0x00020000u   // data_size = 2 (4 bytes)
                    | 0x00100000u   // pad_enable
                    | (6u << 22)    // pad_interval: 128 DWORDs
                    | (3u << 25));  // pad_amount: 4 DWORDs
        g1[1] = (int)(128u << 16);  // tensor_dim0[15:0]=128 (abar addr = 0)
        g1[2] = (int)(128u << 16);  // tensor_dim1[15:0]=128 (dim0 hi = 0)
        g1[3] = (int)(128u << 16);  // tile_dim0=128 (dim1 hi = 0)
        g1[4] = 128;                // tile_dim1=128, tile_dim2=0
        g1[5] = (int)(unsigned)src_row_stride;  // tensor_dim0_stride[31:0]
        g1[6] = 0;                  // stride hi, dim1_stride lo
        g1[7] = 0;

        __builtin_amdgcn_tensor_load_to_lds(g0, g1, (i32x4)0, (i32x4)0,
                                            (i32x8)0, 0);
    }

    // ---- Stage A[n][k] = R[n][k] * (1 - prev_h[b][n]^2) (overlaps the DMA)
    for (int t = tid; t < HDIM * HDIM; t += 256) {
        int n = t >> 7;
        int k = t & 127;
        float p = prev_h[b * HDIM + n];
        a_lds[n * LSTRIDE + k] = R[t] * (1.0f - p * p);
    }

    if (tid < 32) __builtin_amdgcn_s_wait_tensorcnt(0);   // issuing wave drains TDM
    __syncthreads();                                      // publish tile to block

    // ---- WMMA compute: wave w owns k rows [w*16, w*16+16), all 128 columns
    const int wave = tid >> 5;
    const int lane = tid & 31;
    const int half = lane >> 4;    // 0: lanes 0-15, 1: lanes 16-31
    const int lm   = lane & 15;
    const int kk   = wave * 16 + lm;   // A fragment row (M = lane%16)

    v8f acc0 = 0.0f, acc1 = 0.0f, acc2 = 0.0f, acc3 = 0.0f;
    v8f acc4 = 0.0f, acc5 = 0.0f, acc6 = 0.0f, acc7 = 0.0f;

    for (int n0 = 0; n0 < HDIM; n0 += 4) {
        const int r0 = n0 + 2 * half;      // K = vgpr + 2*(lane>=16)
        v2f a;
        a.x = a_lds[r0 * LSTRIDE + kk];
        a.y = a_lds[(r0 + 1) * LSTRIDE + kk];

        const float* brow0 = &s_lds[r0 * LSTRIDE + lm];
        const float* brow1 = &s_lds[(r0 + 1) * LSTRIDE + lm];

        #define WMMA_STEP(CI, ACC)                                              \
        {                                                                       \
            v2f bb;                                                             \
            bb.x = brow0[(CI) * 16];                                            \
            bb.y = brow1[(CI) * 16];                                            \
            ACC = __builtin_amdgcn_wmma_f32_16x16x4_f32(                        \
                false, a, false, bb, (short)0, ACC, false, false);              \
        }
        WMMA_STEP(0, acc0) WMMA_STEP(1, acc1) WMMA_STEP(2, acc2) WMMA_STEP(3, acc3)
        WMMA_STEP(4, acc4) WMMA_STEP(5, acc5) WMMA_STEP(6, acc6) WMMA_STEP(7, acc7)
        #undef WMMA_STEP
    }

    // ---- Epilogue: diagonal add + store. C/D layout: M = v + 8*(lane>=16)
    v8f accs[8] = {acc0, acc1, acc2, acc3, acc4, acc5, acc6, acc7};
    #pragma unroll
    for (int ci = 0; ci < 8; ++ci) {
        const int cl = ci * 16 + lm;       // local column (== j)
        #pragma unroll
        for (int v = 0; v < 8; ++v) {
            const int kg = wave * 16 + v + 8 * half;   // global k row
            float val = accs[ci][v];
            if (cl == kg) val += diag;
            ob[(long long)kg * out_row_stride + cl] = val;
        }
    }
}

extern "C" void kernel_launch(void* const* d_in, const int* in_sizes, int n_in,
                              void* d_out, int out_size, void* d_ws, size_t ws_size,
                              hipStream_t stream) {
    (void)in_sizes; (void)n_in; (void)out_size; (void)d_ws; (void)ws_size;
    const float* x      = (const float*)d_in[0];
    const float* prev_h = (const float*)d_in[1];
    const float* S_R    = (const float*)d_in[2];
    const float* S_W    = (const float*)d_in[3];
    const float* S_B    = (const float*)d_in[4];
    const float* W      = (const float*)d_in[5];
    const float* bias   = (const float*)d_in[6];
    const float* R      = (const float*)d_in[7];
    float* out = (float*)d_out;

    rtrl_h_kernel<<<1, 256, 0, stream>>>(x, prev_h, W, bias, R, out);

    const size_t shmem = (size_t)2 * HDIM * LSTRIDE * sizeof(float); // ~132 KB
    rtrl_sens_kernel<<<BDIM * (2 * HDIM + 1), 256, shmem, stream>>>(
        x, prev_h, S_R, S_W, S_B, R, out);
}